// TensorizedLinear_36790689858242
// MI455X (gfx1250) — compile-verified
//
#include <hip/hip_runtime.h>

// TensorizedLinear (TT-matrix, rank 16) fused forward for MI455X (gfx1250).
// One workgroup (256 threads = 8 wave32) per token. Entire chain fused:
//   gather(input_perm) -> GEMM1 (G1^T contraction, LDS-resident Tt)
//   -> GEMM2 (G0 contraction, persistent WMMA accumulators)
//   -> alpha*per_dim_scale -> gather(output_inv_perm) -> +bias
// All matrix math on v_wmma_f32_16x16x4_f32 (full fp32 fidelity).
// GEMM1 processes c-tile PAIRS per wave so each LDS B operand feeds 2 WMMAs.

#define TOKENS 2048
#define INF 4096
#define OUTF 4096
#define JB 32                 // j-chunk size (2 chunks of 32 j's)
#define NCHUNK 2
#define XP_STRIDE 65          // 64 + 1 pad (bank-conflict avoidance)
#define TT_STRIDE (JB + 1)    // 33, coprime with 64 banks
#define SMEM_FLOATS (1024 * TT_STRIDE + 64 * XP_STRIDE)
#define SMEM_BYTES (SMEM_FLOATS * 4)   // 151,808 B -> 2 WGs per WGP (320 KB)

typedef __attribute__((ext_vector_type(2))) float v2f;
typedef __attribute__((ext_vector_type(8))) float v8f;

__global__ __launch_bounds__(256)
void tt_linear_kernel(const float* __restrict__ x,     // (2048, 4096)
                      const float* __restrict__ g0,    // (1, 64, 64, 16) -> [i*1024 + j*16 + r]
                      const float* __restrict__ g1,    // (16, 64, 64, 1) -> [(r*64+m)*64 + k]
                      const float* __restrict__ alpha, // (1,)
                      const float* __restrict__ pds,   // (4096,)
                      const float* __restrict__ bias,  // (4096,)
                      const int*   __restrict__ iperm, // (4096,)
                      const int*   __restrict__ oinv,  // (4096,)
                      float* __restrict__ out)         // (2048, 4096)
{
    extern __shared__ float smem[];
    float* Tt = smem;                        // [1024][TT_STRIDE]  c=(r*64+m), j-local
    float* xp = smem + 1024 * TT_STRIDE;     // [64][XP_STRIDE]; reused as ylds

    const int b    = blockIdx.x;
    const int tid  = threadIdx.x;
    const int lane = tid & 31;
    const int w    = tid >> 5;
    const int half = lane >> 4;              // 0 | 1
    const int l15  = lane & 15;
    const int koff = half * 2;               // A/B K sub-offset per ISA 16x4 f32 layout

    // ---- gather permuted input row into LDS as xp[j][k], j*64+k = perm index
    const float* xrow = x + (size_t)b * INF;
    for (int c = tid; c < INF; c += 256) {
        xp[(c >> 6) * XP_STRIDE + (c & 63)] = xrow[iperm[c]];
    }
    __syncthreads();

    // ---- persistent Y accumulators: wave w owns Y tiles (ti, tm0) and (ti, tm0+1)
    const int ti  = w & 3;
    const int tm0 = (w >> 2) * 2;
    v8f accY0 = {};
    v8f accY1 = {};

    for (int chunk = 0; chunk < NCHUNK; ++chunk) {
        const int jbase = chunk * JB;

        // ======== GEMM1: Tt[c][jl] = sum_k g1[c*64+k] * xp[jbase+jl][k] ========
        // 64 c-tiles (M) x 2 j-tiles (N), K=64. Wave w handles c-tile pairs
        // {2w, 2w+1} + 16*p so each B (xp) LDS load is shared by 2 WMMAs.
        for (int cp = 0; cp < 4; ++cp) {
            const int ct0   = w * 2 + cp * 16;
            const int cb0   = ct0 * 16;
            const int cb1   = cb0 + 16;
            const float* ap0 = g1 + (cb0 + l15) * 64 + koff;   // A rows, c-tile 0
            const float* ap1 = g1 + (cb1 + l15) * 64 + koff;   // A rows, c-tile 1
            v8f t00 = {}, t01 = {};   // c-tile 0 x {j-tile 0, 1}
            v8f t10 = {}, t11 = {};   // c-tile 1 x {j-tile 0, 1}
            #pragma unroll
            for (int kb = 0; kb < 64; kb += 4) {
                v2f a0 = *(const v2f*)(ap0 + kb);
                v2f a1 = *(const v2f*)(ap1 + kb);
                const int k = kb + koff;
                v2f b0, b1;
                {
                    const float* bp = xp + (jbase + l15) * XP_STRIDE + k;
                    b0.x = bp[0]; b0.y = bp[1];
                }
                {
                    const float* bp = xp + (jbase + 16 + l15) * XP_STRIDE + k;
                    b1.x = bp[0]; b1.y = bp[1];
                }
                t00 = __builtin_amdgcn_wmma_f32_16x16x4_f32(false, a0, false, b0, (short)0, t00, false, false);
                t01 = __builtin_amdgcn_wmma_f32_16x16x4_f32(false, a0, false, b1, (short)0, t01, false, false);
                t10 = __builtin_amdgcn_wmma_f32_16x16x4_f32(false, a1, false, b0, (short)0, t10, false, false);
                t11 = __builtin_amdgcn_wmma_f32_16x16x4_f32(false, a1, false, b1, (short)0, t11, false, false);
            }
            // D layout: VGPR v -> row v + 8*half, col = l15
            #pragma unroll
            for (int v = 0; v < 8; ++v) {
                const int r0 = cb0 + v + half * 8;
                const int r1 = cb1 + v + half * 8;
                Tt[r0 * TT_STRIDE + l15]      = t00[v];
                Tt[r0 * TT_STRIDE + 16 + l15] = t01[v];
                Tt[r1 * TT_STRIDE + l15]      = t10[v];
                Tt[r1 * TT_STRIDE + 16 + l15] = t11[v];
            }
        }
        __syncthreads();

        // ======== GEMM2: Y[i][m] += sum_kk g0[i*1024 + jbase*16 + kk] * Tt[(kk&15)*64+m][kk>>4]
        // kk enumerates (j_local, r) pairs: kk = jl*16 + r, K-chunk = 512.
        const float* aptr = g0 + (ti * 16 + l15) * 1024 + jbase * 16 + koff;
        #pragma unroll 4
        for (int ks = 0; ks < JB * 16; ks += 4) {
            v2f a = *(const v2f*)(aptr + ks);
            const int kk = ks + koff;        // ks%16 in {0,4,8,12}, koff in {0,2}: no j wrap
            const int r  = kk & 15;
            const int jl = kk >> 4;
            {
                const int m = tm0 * 16 + l15;
                v2f bv;
                bv.x = Tt[(r * 64 + m) * TT_STRIDE + jl];
                bv.y = Tt[((r + 1) * 64 + m) * TT_STRIDE + jl];
                accY0 = __builtin_amdgcn_wmma_f32_16x16x4_f32(false, a, false, bv, (short)0, accY0, false, false);
            }
            {
                const int m = (tm0 + 1) * 16 + l15;
                v2f bv;
                bv.x = Tt[(r * 64 + m) * TT_STRIDE + jl];
                bv.y = Tt[((r + 1) * 64 + m) * TT_STRIDE + jl];
                accY1 = __builtin_amdgcn_wmma_f32_16x16x4_f32(false, a, false, bv, (short)0, accY1, false, false);
            }
        }
        __syncthreads();   // Tt fully consumed before next chunk overwrites it
    }

    // ---- epilogue: y_pre * alpha * per_dim_scale staged in LDS (reuse xp buffer)
    const float av = alpha[0];
    float* ylds = xp;
    #pragma unroll
    for (int v = 0; v < 8; ++v) {
        const int i  = ti * 16 + v + half * 8;
        const int m0 = tm0 * 16 + l15;
        const int m1 = m0 + 16;
        ylds[i * XP_STRIDE + m0] = accY0[v] * av * pds[i * 64 + m0];
        ylds[i * XP_STRIDE + m1] = accY1[v] * av * pds[i * 64 + m1];
    }
    __syncthreads();

    // ---- output inverse permutation gather + bias
    float* orow = out + (size_t)b * OUTF;
    for (int q = tid; q < OUTF; q += 256) {
        const int o = oinv[q];
        orow[q] = ylds[(o >> 6) * XP_STRIDE + (o & 63)] + bias[q];
    }
}

extern "C" void kernel_launch(void* const* d_in, const int* in_sizes, int n_in,
                              void* d_out, int out_size, void* d_ws, size_t ws_size,
                              hipStream_t stream) {
    (void)in_sizes; (void)n_in; (void)out_size; (void)d_ws; (void)ws_size;
    const float* x     = (const float*)d_in[0];
    const float* g0    = (const float*)d_in[1];
    const float* g1    = (const float*)d_in[2];
    const float* alpha = (const float*)d_in[3];
    const float* pds   = (const float*)d_in[4];
    const float* bias  = (const float*)d_in[5];
    const int*   iperm = (const int*)d_in[6];
    const int*   oinv  = (const int*)d_in[7];
    float* out = (float*)d_out;

    // allow >64KB dynamic LDS (idempotent, not a stream op: graph-capture safe)
    (void)hipFuncSetAttribute(reinterpret_cast<const void*>(tt_linear_kernel),
                              hipFuncAttributeMaxDynamicSharedMemorySize, SMEM_BYTES);

    tt_linear_kernel<<<TOKENS, 256, SMEM_BYTES, stream>>>(
        x, g0, g1, alpha, pds, bias, iperm, oinv, out);
}